// DeformAlign_17463337025963
// MI455X (gfx1250) — compile-verified
//
#include <hip/hip_runtime.h>
#include <hip/hip_bf16.h>

typedef __attribute__((ext_vector_type(2))) float v2f;
typedef __attribute__((ext_vector_type(4))) float v4f;
typedef __attribute__((ext_vector_type(8))) float v8f;

#define Bn 8
#define Cc 64
#define Hh 128
#define Ww 128
#define HW (Hh * Ww)   // 16384 = 2^14

__device__ __forceinline__ v8f wmma_f32x4(v2f a, v2f b, v8f c) {
  // D = A(16x4 f32) * B(4x16 f32) + C(16x16 f32), full fp32 precision
  return __builtin_amdgcn_wmma_f32_16x16x4_f32(false, a, false, b, (short)0, c,
                                               false, false);
}

// ---------------------------------------------------------------------------
// NCHW [8][64][HW] -> NHWC [8][HW][64]  (channels contiguous -> b64 loads)
// ---------------------------------------------------------------------------
__global__ void nchw2nhwc_k(const float* __restrict__ in,
                            float* __restrict__ out) {
  const size_t i = (size_t)blockIdx.x * blockDim.x + threadIdx.x;
  const int c = (int)(i & 63);
  const size_t bp = i >> 6;          // b*HW + p
  const int p = (int)(bp & (HW - 1));
  const int b = (int)(bp >> 14);
  out[i] = in[((size_t)(b * 64 + c)) * HW + p];
}

// ---------------------------------------------------------------------------
// Weight repack into per-lane WMMA A-fragments:
//   wF[tap][chunk][ot][lane] = { w[o][ca][tap], w[o][ca+1][tap] }
// with o = ot*16 + (lane&15), ca = chunk*4 + 2*(lane>>4); zero-pad o >= O.
// ---------------------------------------------------------------------------
__global__ void repack_wfrag_k(const float* __restrict__ w,
                               v2f* __restrict__ wF, int O, int Ot) {
  const int i = blockIdx.x * blockDim.x + threadIdx.x;
  const int total = 9 * 16 * Ot * 32;
  if (i >= total) return;
  const int lane = i & 31;
  const int ot = (i >> 5) % Ot;
  const int chunk = (i / (32 * Ot)) % 16;
  const int tap = i / (32 * Ot * 16);
  const int n = lane & 15, half = lane >> 4;
  const int ca = chunk * 4 + 2 * half;
  const int o = ot * 16 + n;
  v2f v;
  v.x = (o < O) ? w[o * (Cc * 9) + ca * 9 + tap] : 0.0f;
  v.y = (o < O) ? w[o * (Cc * 9) + (ca + 1) * 9 + tap] : 0.0f;
  wF[i] = v;
}

// ---------------------------------------------------------------------------
// Kernel A: conv1 (64->64, 3x3, pad 1) + ReLU.  NHWC in/out.
// A-fragments staged per-tap in LDS (double-buffered); hot loop:
// 2 b64 VMEM B-loads + 4 b64 DS A-loads -> 8 WMMAs per chunk.
// ---------------------------------------------------------------------------
__global__ __launch_bounds__(256) void conv1_relu_k(
    const float* __restrict__ refT, const v2f* __restrict__ wF,
    const float* __restrict__ b1, float* __restrict__ outT) {
  __shared__ v2f sWF[2][2048];  // 2 x 16KB: one tap slice [chunk][ot][lane]
  const int lane = threadIdx.x & 31;
  const int wave = blockIdx.x * 8 + (threadIdx.x >> 5);
  const int b    = wave >> 9;
  const int rem  = wave & 511;
  const int h    = rem >> 2;
  const int w0   = (rem & 3) << 5;
  const int half = lane >> 4;
  const int n    = lane & 15;

  // stage tap 0
#pragma unroll
  for (int k = 0; k < 8; ++k) sWF[0][threadIdx.x + 256 * k] = wF[threadIdx.x + 256 * k];
  __syncthreads();

  v8f acc[4][2];
  v8f zero = {};
#pragma unroll
  for (int i = 0; i < 4; ++i) {
    acc[i][0] = zero;
    acc[i][1] = zero;
  }

  const float* refb = refT + ((size_t)b * HW) * 64;
#pragma unroll
  for (int tap = 0; tap < 9; ++tap) {
    const int cur = tap & 1;
    if (tap + 1 < 9) {  // stage next tap into the other buffer
      const v2f* src = wF + (tap + 1) * 2048;
#pragma unroll
      for (int k = 0; k < 8; ++k)
        sWF[cur ^ 1][threadIdx.x + 256 * k] = src[threadIdx.x + 256 * k];
    }
    const int ki = tap / 3, kj = tap % 3;
    const int yy = h + ki - 1;
    const float rmsk = (yy >= 0 && yy < Hh) ? 1.0f : 0.0f;
    const int yyc = min(max(yy, 0), Hh - 1);
    float msk[2];
    const float* rp[2];
#pragma unroll
    for (int t = 0; t < 2; ++t) {
      const int xx = w0 + 16 * t + n + kj - 1;
      msk[t] = (xx >= 0 && xx < Ww) ? rmsk : 0.0f;
      const int xxc = min(max(xx, 0), Ww - 1);
      rp[t] = refb + (size_t)(yyc * Ww + xxc) * 64;
    }
    for (int chunk = 0; chunk < 16; ++chunk) {
      const int ca = chunk * 4 + 2 * half;
      v2f bf[2];
#pragma unroll
      for (int t = 0; t < 2; ++t)
        bf[t] = *(const v2f*)(rp[t] + ca) * msk[t];
      v2f af[4];
#pragma unroll
      for (int ot = 0; ot < 4; ++ot)
        af[ot] = sWF[cur][(chunk * 4 + ot) * 32 + lane];
#pragma unroll
      for (int ot = 0; ot < 4; ++ot) {
        acc[ot][0] = wmma_f32x4(af[ot], bf[0], acc[ot][0]);
        acc[ot][1] = wmma_f32x4(af[ot], bf[1], acc[ot][1]);
      }
    }
    __syncthreads();  // staged buffer ready / old buffer free
  }
  // NHWC store: 8 consecutive oc per lane -> two b128 stores per (ot, tile)
#pragma unroll
  for (int t = 0; t < 2; ++t) {
    float* tb = outT + ((size_t)b * HW + h * Ww + w0 + 16 * t + n) * 64;
#pragma unroll
    for (int ot = 0; ot < 4; ++ot) {
      float* p0 = tb + ot * 16 + 8 * half;
      const int ob = ot * 16 + 8 * half;
      v4f lo, hi;
#pragma unroll
      for (int r = 0; r < 4; ++r) {
        float v = acc[ot][t][r] + b1[ob + r];
        lo[r] = v > 0.0f ? v : 0.0f;
      }
#pragma unroll
      for (int r = 0; r < 4; ++r) {
        float v = acc[ot][t][4 + r] + b1[ob + 4 + r];
        hi[r] = v > 0.0f ? v : 0.0f;
      }
      *(v4f*)(p0) = lo;
      *(v4f*)(p0 + 4) = hi;
    }
  }
}

// ---------------------------------------------------------------------------
// Kernel B: conv2 (64->18) -> offsets [b][p][18].  Same LDS staging (8KB x2).
// ---------------------------------------------------------------------------
__global__ __launch_bounds__(256) void conv2_k(
    const float* __restrict__ inT, const v2f* __restrict__ wF,
    const float* __restrict__ b2, float* __restrict__ offsT) {
  __shared__ v2f sWF[2][1024];
  const int lane = threadIdx.x & 31;
  const int wave = blockIdx.x * 8 + (threadIdx.x >> 5);
  const int b    = wave >> 9;
  const int rem  = wave & 511;
  const int h    = rem >> 2;
  const int w0   = (rem & 3) << 5;
  const int half = lane >> 4;
  const int n    = lane & 15;

#pragma unroll
  for (int k = 0; k < 4; ++k) sWF[0][threadIdx.x + 256 * k] = wF[threadIdx.x + 256 * k];
  __syncthreads();

  v8f acc[2][2];
  v8f zero = {};
#pragma unroll
  for (int i = 0; i < 2; ++i) {
    acc[i][0] = zero;
    acc[i][1] = zero;
  }

  const float* inb = inT + ((size_t)b * HW) * 64;
#pragma unroll
  for (int tap = 0; tap < 9; ++tap) {
    const int cur = tap & 1;
    if (tap + 1 < 9) {
      const v2f* src = wF + (tap + 1) * 1024;
#pragma unroll
      for (int k = 0; k < 4; ++k)
        sWF[cur ^ 1][threadIdx.x + 256 * k] = src[threadIdx.x + 256 * k];
    }
    const int ki = tap / 3, kj = tap % 3;
    const int yy = h + ki - 1;
    const float rmsk = (yy >= 0 && yy < Hh) ? 1.0f : 0.0f;
    const int yyc = min(max(yy, 0), Hh - 1);
    float msk[2];
    const float* rp[2];
#pragma unroll
    for (int t = 0; t < 2; ++t) {
      const int xx = w0 + 16 * t + n + kj - 1;
      msk[t] = (xx >= 0 && xx < Ww) ? rmsk : 0.0f;
      const int xxc = min(max(xx, 0), Ww - 1);
      rp[t] = inb + (size_t)(yyc * Ww + xxc) * 64;
    }
    for (int chunk = 0; chunk < 16; ++chunk) {
      const int ca = chunk * 4 + 2 * half;
      v2f bf[2];
#pragma unroll
      for (int t = 0; t < 2; ++t)
        bf[t] = *(const v2f*)(rp[t] + ca) * msk[t];
      v2f af[2];
#pragma unroll
      for (int ot = 0; ot < 2; ++ot)
        af[ot] = sWF[cur][(chunk * 2 + ot) * 32 + lane];
#pragma unroll
      for (int ot = 0; ot < 2; ++ot) {
        acc[ot][0] = wmma_f32x4(af[ot], bf[0], acc[ot][0]);
        acc[ot][1] = wmma_f32x4(af[ot], bf[1], acc[ot][1]);
      }
    }
    __syncthreads();
  }
  // Store channels 0..17 of [b][p][18] as b64 pairs.
#pragma unroll
  for (int t = 0; t < 2; ++t) {
    float* ob = offsT + ((size_t)b * HW + h * Ww + w0 + 16 * t + n) * 18;
    float* p0 = ob + 8 * half;
#pragma unroll
    for (int rr = 0; rr < 4; ++rr) {
      const int o = 8 * half + 2 * rr;
      v2f s;
      s.x = acc[0][t][2 * rr] + b2[o];
      s.y = acc[0][t][2 * rr + 1] + b2[o + 1];
      *(v2f*)(p0 + 2 * rr) = s;
    }
    if (half == 0) {  // rows 16,17 live in ot=1, lanes 0-15
      v2f s;
      s.x = acc[1][t][0] + b2[16];
      s.y = acc[1][t][1] + b2[17];
      *(v2f*)(ob + 16) = s;
    }
  }
}

// ---------------------------------------------------------------------------
// Kernel C: deformable conv.  x NHWC, offsets [b][p][18], out NCHW.
// Hot loop: 8 b64 VMEM corner loads + 4 b64 DS A-loads -> 8 WMMAs per chunk.
// ---------------------------------------------------------------------------
__global__ __launch_bounds__(256) void deform_k(
    const float* __restrict__ xT, const float* __restrict__ offsT,
    const v2f* __restrict__ wF, float* __restrict__ out) {
  __shared__ v2f sWF[2][2048];
  const int lane = threadIdx.x & 31;
  const int wave = blockIdx.x * 8 + (threadIdx.x >> 5);
  const int b    = wave >> 9;
  const int rem  = wave & 511;
  const int h    = rem >> 2;
  const int w0   = (rem & 3) << 5;
  const int half = lane >> 4;
  const int n    = lane & 15;

#pragma unroll
  for (int k = 0; k < 8; ++k) sWF[0][threadIdx.x + 256 * k] = wF[threadIdx.x + 256 * k];
  __syncthreads();

  v8f acc[4][2];
  v8f zero = {};
#pragma unroll
  for (int i = 0; i < 4; ++i) {
    acc[i][0] = zero;
    acc[i][1] = zero;
  }

  const float* xb = xT + ((size_t)b * HW) * 64;

#pragma unroll
  for (int tap = 0; tap < 9; ++tap) {
    const int cur = tap & 1;
    if (tap + 1 < 9) {
      const v2f* src = wF + (tap + 1) * 2048;
#pragma unroll
      for (int k = 0; k < 8; ++k)
        sWF[cur ^ 1][threadIdx.x + 256 * k] = src[threadIdx.x + 256 * k];
    }
    const int ki = tap / 3, kj = tap % 3;
    const float* cbase[2][4];
    float cw[2][4];
#pragma unroll
    for (int t = 0; t < 2; ++t) {
      const int w = w0 + 16 * t + n;
      const v2f oyx = *(const v2f*)(offsT +
                                    ((size_t)b * HW + h * Ww + w) * 18 +
                                    2 * tap);
      const float ys = oyx.x + (float)(h + ki - 1);
      const float xs = oyx.y + (float)(w + kj - 1);
      const float y0f = floorf(ys), x0f = floorf(xs);
      const float dy = ys - y0f, dx = xs - x0f;
      const int y0 = (int)y0f, x0 = (int)x0f;
#pragma unroll
      for (int i = 0; i < 2; ++i)
#pragma unroll
        for (int j = 0; j < 2; ++j) {
          const int yy = y0 + i, xx = x0 + j;
          const float v =
              (yy >= 0 && yy < Hh && xx >= 0 && xx < Ww) ? 1.0f : 0.0f;
          const int yyc = min(max(yy, 0), Hh - 1);
          const int xxc = min(max(xx, 0), Ww - 1);
          cbase[t][i * 2 + j] = xb + (size_t)(yyc * Ww + xxc) * 64;
          cw[t][i * 2 + j] = (i ? dy : 1.0f - dy) * (j ? dx : 1.0f - dx) * v;
        }
    }
    for (int chunk = 0; chunk < 16; ++chunk) {
      const int ca = chunk * 4 + 2 * half;
      v2f bf[2];
#pragma unroll
      for (int t = 0; t < 2; ++t) {
        const v2f c0 = *(const v2f*)(cbase[t][0] + ca);
        const v2f c1 = *(const v2f*)(cbase[t][1] + ca);
        const v2f c2 = *(const v2f*)(cbase[t][2] + ca);
        const v2f c3 = *(const v2f*)(cbase[t][3] + ca);
        bf[t] = c0 * cw[t][0] + c1 * cw[t][1] + c2 * cw[t][2] + c3 * cw[t][3];
      }
      v2f af[4];
#pragma unroll
      for (int ot = 0; ot < 4; ++ot)
        af[ot] = sWF[cur][(chunk * 4 + ot) * 32 + lane];
#pragma unroll
      for (int ot = 0; ot < 4; ++ot) {
        acc[ot][0] = wmma_f32x4(af[ot], bf[0], acc[ot][0]);
        acc[ot][1] = wmma_f32x4(af[ot], bf[1], acc[ot][1]);
      }
    }
    __syncthreads();
  }
  // Output must be NCHW: coalesced b32 stores across lanes.
  float* op = out + (size_t)b * 64 * HW + h * Ww + w0 + n;
#pragma unroll
  for (int ot = 0; ot < 4; ++ot)
#pragma unroll
    for (int r = 0; r < 8; ++r) {
      const int o = ot * 16 + r + 8 * half;
#pragma unroll
      for (int t = 0; t < 2; ++t)
        op[(size_t)o * HW + 16 * t] = acc[ot][t][r];
    }
}

// ---------------------------------------------------------------------------
extern "C" void kernel_launch(void* const* d_in, const int* in_sizes, int n_in,
                              void* d_out, int out_size, void* d_ws,
                              size_t ws_size, hipStream_t stream) {
  const float* x   = (const float*)d_in[0];
  const float* rf  = (const float*)d_in[1];
  const float* w1  = (const float*)d_in[2];
  const float* b1  = (const float*)d_in[3];
  const float* w2  = (const float*)d_in[4];
  const float* b2  = (const float*)d_in[5];
  const float* wt  = (const float*)d_in[6];
  float* out = (float*)d_out;

  const size_t act = (size_t)Bn * HW * 64;   // 8.39M floats
  float* actT  = (float*)d_ws;               // shared: refT then xT
  float* tmp1  = actT + act;                 // conv1 out, NHWC
  float* offsT = tmp1 + act;                 // [8][HW][18]
  v2f*   wF1   = (v2f*)(offsT + (size_t)Bn * HW * 18);  // [9][16][4][32]
  v2f*   wF2   = wF1 + 9 * 16 * 4 * 32;                 // [9][16][2][32]
  v2f*   wFd   = wF2 + 9 * 16 * 2 * 32;                 // [9][16][4][32]

  repack_wfrag_k<<<(9 * 16 * 4 * 32 + 255) / 256, 256, 0, stream>>>(w1, wF1,
                                                                    64, 4);
  repack_wfrag_k<<<(9 * 16 * 2 * 32 + 255) / 256, 256, 0, stream>>>(w2, wF2,
                                                                    18, 2);
  repack_wfrag_k<<<(9 * 16 * 4 * 32 + 255) / 256, 256, 0, stream>>>(wt, wFd,
                                                                    64, 4);

  const int tgrid = (int)(act / 256);  // 32768 blocks
  dim3 grid(512), block(256);          // 4096 wave32 tiles of 32 pixels

  nchw2nhwc_k<<<tgrid, 256, 0, stream>>>(rf, actT);          // refT
  conv1_relu_k<<<grid, block, 0, stream>>>(actT, wF1, b1, tmp1);
  nchw2nhwc_k<<<tgrid, 256, 0, stream>>>(x, actT);           // reuse as xT
  conv2_k<<<grid, block, 0, stream>>>(tmp1, wF2, b2, offsT);
  deform_k<<<grid, block, 0, stream>>>(actT, offsT, wFd, out);
}